// ConvLSTMCell4_41420664602680
// MI455X (gfx1250) — compile-verified
//
#include <hip/hip_runtime.h>
#include <stdint.h>

#define NGROUPS 19
#define HH 256
#define WD 512
#define TILE_H 16
#define TILE_W 64
#define LDSH 18
#define LDSW 72                  /* padded stride: core at col 4 -> 16B aligned rows */
#define PLANE (LDSH * LDSW)      /* 1296 floats */
#define IN_PLANES 6              /* 5 real + 1 zero plane for K padding */
#define KPAD 48                  /* 45 real K padded to 12 chunks of 4 */
#define CORE_COL 4               /* lds column of gx == bx0 */

typedef __attribute__((ext_vector_type(2))) float v2f;
typedef __attribute__((ext_vector_type(8))) float v8f;

__device__ __forceinline__ float fast_sigmoid(float x) {
    return 1.0f / (1.0f + __expf(-x));
}
__device__ __forceinline__ float fast_tanh(float x) {
    return 1.0f - 2.0f / (__expf(2.0f * x) + 1.0f);
}

__global__ __launch_bounds__(256) void convlstm_wmma_kernel(
    const float* __restrict__ xin,   // (B,19,H,W)
    const float* __restrict__ hin,   // (B,19,4,H,W)
    const float* __restrict__ cin,   // (B,19,4,H,W)
    const float* __restrict__ Wt,    // (304,5,3,3)
    const float* __restrict__ bias,  // (304,)
    float* __restrict__ out_h,       // (B,19,4,H,W)
    float* __restrict__ out_c)       // (B,19,4,H,W)
{
    __shared__ __align__(16) float S[IN_PLANES * PLANE];  // planes 0..4 real, plane 5 stays zero
    __shared__ __align__(16) float Wl[16 * KPAD];         // group weights 16 x 48 (K-padded)

    const int tid = threadIdx.x;
    const int bg  = blockIdx.z;             // b*19 + g
    const int g   = bg % NGROUPS;
    const int by0 = blockIdx.y * TILE_H;
    const int bx0 = blockIdx.x * TILE_W;

    // ---- zero-fill LDS (halo padding + pad columns + zero plane + K padding) ----
    for (int e = tid; e < IN_PLANES * PLANE; e += 256) S[e] = 0.0f;
    for (int e = tid; e < 16 * KPAD; e += 256) Wl[e] = 0.0f;
    __syncthreads();

    // ---- stage per-group weights: Wl[m][k] = W[g*16+m][k], k = ci*9 + ky*3 + kx ----
    const float* Wg = Wt + (size_t)g * 16 * 45;
    for (int e = tid; e < 16 * 45; e += 256) {
        int m = e / 45, k = e - m * 45;
        Wl[m * KPAD + k] = Wg[e];
    }

    // ---- async-stage 5 input planes (x, h0..h3) into LDS ----
    const float* xplane = xin + (size_t)bg * HH * WD;
    const float* hbase  = hin + (size_t)bg * 4 * HH * WD;

    // core: 16B-aligned B128 transfers, 16 per row per plane
    for (int e = tid; e < 5 * LDSH * 16; e += 256) {
        int ci = e / (LDSH * 16);
        int r  = e - ci * (LDSH * 16);
        int ly = r >> 4;
        int t  = r & 15;
        int gy = by0 - 1 + ly;
        if ((unsigned)gy < (unsigned)HH) {
            int gx = bx0 + 4 * t;
            const float* gp = (ci == 0)
                ? (xplane + (size_t)gy * WD + gx)
                : (hbase + ((size_t)(ci - 1) * HH + gy) * WD + gx);
            uint32_t lo = (uint32_t)(uintptr_t)(&S[ci * PLANE + ly * LDSW + CORE_COL + 4 * t]);
            asm volatile("global_load_async_to_lds_b128 %0, %1, off"
                         :: "v"(lo), "v"(gp) : "memory");
        }
    }
    // halo columns (left: gx = bx0-1 -> col 3, right: gx = bx0+64 -> col 68)
    for (int e = tid; e < 5 * LDSH * 2; e += 256) {
        int ci   = e / (LDSH * 2);
        int r    = e - ci * (LDSH * 2);
        int ly   = r >> 1;
        int side = r & 1;
        int gy = by0 - 1 + ly;
        int gx = side ? (bx0 + TILE_W) : (bx0 - 1);
        if ((unsigned)gy < (unsigned)HH && (unsigned)gx < (unsigned)WD) {
            const float* gp = (ci == 0)
                ? (xplane + (size_t)gy * WD + gx)
                : (hbase + ((size_t)(ci - 1) * HH + gy) * WD + gx);
            uint32_t lo = (uint32_t)(uintptr_t)(&S[ci * PLANE + ly * LDSW + (side ? 68 : 3)]);
            asm volatile("global_load_async_to_lds_b32 %0, %1, off"
                         :: "v"(lo), "v"(gp) : "memory");
        }
    }
    asm volatile("s_wait_asynccnt 0x0" ::: "memory");
    __syncthreads();

    // ---- per-lane WMMA fragment setup ----
    const int lane = tid & 31;
    const int wave = tid >> 5;
    const int half = lane >> 4;   // 0: lanes 0-15, 1: lanes 16-31
    const int p    = lane & 15;   // M index for A, N (pixel) index for B

    // A fragments (16x4 f32): lanes 0-15 hold M=lane {K=ks,ks+1}; lanes 16-31 {K=ks+2,ks+3}
    v2f afrag[12];
    int d0[12], d1[12];           // per-lane LDS float-offsets for B gathers
#pragma unroll
    for (int kk = 0; kk < 12; ++kk) {
        int ks = kk * 4 + 2 * half;
        afrag[kk].x = Wl[p * KPAD + ks];
        afrag[kk].y = Wl[p * KPAD + ks + 1];
        int k  = ks;
        int ci = k / 9, tap = k - ci * 9;
        d0[kk] = ci * PLANE + (tap / 3) * LDSW + (tap % 3) + p;
        k  = ks + 1;
        ci = k / 9; tap = k - ci * 9;
        d1[kk] = ci * PLANE + (tap / 3) * LDSW + (tap % 3) + p;
    }

    // bias for the 8 channels this lane accumulates (channel m = v + 8*half)
    float bias_r[8];
#pragma unroll
    for (int v = 0; v < 8; ++v) bias_r[v] = bias[g * 16 + v + 8 * half];

    // ---- 64 segments of 16 pixels; 8 per wave ----
    for (int s = wave; s < 64; s += 8) {
        int sy = s >> 2, sx = s & 3;
        int segbase = sy * LDSW + sx * 16 + (CORE_COL - 1);  // tap dx=0 reads gx-1

        // two independent accumulator chains; chain 0 is seeded with the bias
        v8f acc0, acc1;
#pragma unroll
        for (int v = 0; v < 8; ++v) { acc0[v] = bias_r[v]; acc1[v] = 0.0f; }

#pragma unroll
        for (int kk = 0; kk < 12; kk += 2) {
            v2f b0, b1;
            b0.x = S[segbase + d0[kk]];
            b0.y = S[segbase + d1[kk]];
            b1.x = S[segbase + d0[kk + 1]];
            b1.y = S[segbase + d1[kk + 1]];
            acc0 = __builtin_amdgcn_wmma_f32_16x16x4_f32(
                false, afrag[kk], false, b0, (short)0, acc0, false, false);
            acc1 = __builtin_amdgcn_wmma_f32_16x16x4_f32(
                false, afrag[kk + 1], false, b1, (short)0, acc1, false, false);
        }

        v8f acc;
#pragma unroll
        for (int v = 0; v < 8; ++v) acc[v] = acc0[v] + acc1[v];

        // Exchange accumulator halves: lane pair (p, p+16) swap so each lane
        // sees all four gates for its pixel.
        float r[8];
#pragma unroll
        for (int v = 0; v < 8; ++v) r[v] = __shfl_xor(acc[v], 16, 32);

        int y  = by0 + sy;
        int xg = bx0 + sx * 16 + p;

        // half0 lanes own slabs {0,1}: local = (i,f), remote = (o,g)
        // half1 lanes own slabs {2,3}: local = (o,g), remote = (i,f)
#pragma unroll
        for (int sl = 0; sl < 2; ++sl) {
            float Ai = half ? r[2 + sl]   : acc[sl];
            float Af = half ? r[6 + sl]   : acc[4 + sl];
            float Ao = half ? acc[2 + sl] : r[sl];
            float Ag = half ? acc[6 + sl] : r[4 + sl];
            int slab = 2 * half + sl;

            size_t base = (((size_t)bg * 4 + slab) * HH + y) * WD + xg;
            float cv = cin[base];
            float i_ = fast_sigmoid(Ai);
            float f_ = fast_sigmoid(Af);
            float o_ = fast_sigmoid(Ao);
            float g_ = fast_tanh(Ag);
            float nc = f_ * cv + i_ * g_;
            float nh = o_ * fast_tanh(nc);
            out_h[base] = nh;
            out_c[base] = nc;
        }
    }
}

extern "C" void kernel_launch(void* const* d_in, const int* in_sizes, int n_in,
                              void* d_out, int out_size, void* d_ws, size_t ws_size,
                              hipStream_t stream) {
    (void)in_sizes; (void)n_in; (void)out_size; (void)d_ws; (void)ws_size;
    const float* x = (const float*)d_in[0];
    const float* h = (const float*)d_in[1];
    const float* c = (const float*)d_in[2];
    const float* W = (const float*)d_in[3];
    const float* b = (const float*)d_in[4];
    float* outh = (float*)d_out;
    float* outc = outh + (size_t)4 * NGROUPS * 4 * HH * WD;  // new_h first, then new_c

    dim3 grid(WD / TILE_W, HH / TILE_H, 4 * NGROUPS);
    convlstm_wmma_kernel<<<grid, 256, 0, stream>>>(x, h, c, W, b, outh, outc);
}